// ConvCurv_53532472377497
// MI455X (gfx1250) — compile-verified
//
#include <hip/hip_runtime.h>
#include <math.h>

#define N_NODES 100000
#define N_EDGES 3200000
#define F_IN 256
#define HID 64
#define N_CLS 16

typedef float v2f __attribute__((ext_vector_type(2)));
typedef float v8f __attribute__((ext_vector_type(8)));

#define INF_F __builtin_inff()

// ---------- float atomic max/min via int/uint punning (sign-split trick) ----------
__device__ __forceinline__ void atomicMaxF(float* addr, float v) {
  if (v >= 0.f) atomicMax((int*)addr, __float_as_int(v));
  else          atomicMin((unsigned int*)addr, __float_as_uint(v));
}
__device__ __forceinline__ void atomicMinF(float* addr, float v) {
  if (v >= 0.f) atomicMin((int*)addr, __float_as_int(v));
  else          atomicMax((unsigned int*)addr, __float_as_uint(v));
}

// segment max of z = w*g over the segment, reconstructed from 4 per-node scalars
// ws4 = { max(w>=0), min(w>=0), max(w<0), min(w<0) }  (init -inf,+inf,-inf,+inf)
__device__ __forceinline__ float seg_zmax(const float* ws4, float gpc, float gnc) {
  float wpm = ws4[0], wpn = ws4[1], wnm = ws4[2], wnn = ws4[3];
  float zm = -INF_F;
  if (wpm > -INF_F) zm = fmaxf(zm, (gpc >= 0.f ? wpm : wpn) * gpc);
  if (wnm > -INF_F) zm = fmaxf(zm, (gnc >= 0.f ? wnm : wnn) * gnc);
  return zm;
}

// ---------- tiny utility kernels ----------
__global__ void fill_f32(float* __restrict__ p, float v, long long n) {
  long long i = blockIdx.x * (long long)blockDim.x + threadIdx.x;
  if (i < n) p[i] = v;
}

__global__ void init_wstats(float* __restrict__ w, int n) {
  int i = blockIdx.x * blockDim.x + threadIdx.x;
  if (i < n) {
    float* b = w + 4ll * i;
    b[0] = -INF_F; b[1] = INF_F; b[2] = -INF_F; b[3] = INF_F;
  }
}

// gp[c] = sum_k m0[k]*m1[c,k]*(m0[k]>0 ? 1 : 0.2)   (sign of z when w>0)
// gn[c] = sum_k m0[k]*m1[c,k]*(m0[k]<0 ? 1 : 0.2)   (sign of z when w<0)
__global__ void gvec_kernel(const float* __restrict__ m0, const float* __restrict__ m1,
                            float* __restrict__ gp, float* __restrict__ gn, int C, int K) {
  int c = threadIdx.x;
  if (c >= C) return;
  float sp = 0.f, sn = 0.f;
  for (int k = 0; k < K; ++k) {
    float a = m0[k];
    float prod = a * m1[c * K + k];
    sp += prod * (a > 0.f ? 1.0f : 0.2f);
    sn += prod * (a < 0.f ? 1.0f : 0.2f);
  }
  gp[c] = sp; gn[c] = sn;
}

// ---------- WMMA f32 GEMM:  Out[M,Cout] = A[M,K] @ W[Cout,K]^T + bias ----------
// One wave produces one 16x16 output tile via V_WMMA_F32_16X16X4_F32 over K/4 steps.
// A fragment (16x4 f32): lane<16 -> row=lane, K pair {0,1}; lane>=16 -> row=lane-16, K pair {2,3}
// B fragment (4x16 f32): B[k,n] = W[n,k]; lane<16 -> col n=lane, K pair {0,1}; lane>=16 -> K pair {2,3}
// D (16x16 f32, 8 VGPRs): vgpr r: lanes 0-15 -> (M=r, N=lane); lanes 16-31 -> (M=8+r, N=lane-16)
__global__ void gemm_wmma_f32(const float* __restrict__ A, const float* __restrict__ W,
                              const float* __restrict__ bias, float* __restrict__ Out,
                              int M, int K, int Cout) {
  const int lane = threadIdx.x & 31;
  const int wave = (int)((blockIdx.x * blockDim.x + threadIdx.x) >> 5);
  const int nT = Cout >> 4;
  const int mTile = wave / nT;
  const int nTile = wave % nT;
  if (mTile * 16 >= M) return;          // uniform per wave -> EXEC stays all-ones
  const int half = lane >> 4;
  const int l15  = lane & 15;

  const float* aRow = A + (size_t)(mTile * 16 + l15) * K + half * 2;
  const float* bRow = W + (size_t)(nTile * 16 + l15) * K + half * 2;

  v8f acc = {0.f, 0.f, 0.f, 0.f, 0.f, 0.f, 0.f, 0.f};
#if __has_builtin(__builtin_amdgcn_wmma_f32_16x16x4_f32)
  #pragma unroll 8
  for (int k = 0; k < K; k += 4) {
    v2f a = *(const v2f*)(aRow + k);
    v2f b = *(const v2f*)(bRow + k);
    acc = __builtin_amdgcn_wmma_f32_16x16x4_f32(false, a, false, b, (short)0, acc, false, false);
  }
#else
  // scalar fallback matching the D layout
  for (int r = 0; r < 8; ++r) {
    int row = mTile * 16 + r + half * 8;
    const float* ar = A + (size_t)row * K;
    const float* wr = W + (size_t)(nTile * 16 + l15) * K;
    float s = 0.f;
    for (int k = 0; k < K; ++k) s += ar[k] * wr[k];
    acc[r] = s;
  }
#endif
  float bc = bias[nTile * 16 + l15];
  #pragma unroll
  for (int r = 0; r < 8; ++r) {
    int row = mTile * 16 + r + half * 8;
    Out[(size_t)row * Cout + (nTile * 16 + l15)] = acc[r] + bc;
  }
}

// ---------- edge passes ----------
__global__ void edge_minmax(const int* __restrict__ ei, const float* __restrict__ wm,
                            float* __restrict__ wstats) {
  int e = blockIdx.x * blockDim.x + threadIdx.x;
  if (e >= N_EDGES) return;
  int s = ei[e];                 // src = edge_index[0]
  float w = wm[e];
  float* b = wstats + 4ll * s;
  if (w >= 0.f) { atomicMaxF(b + 0, w); atomicMinF(b + 1, w); }
  else          { atomicMaxF(b + 2, w); atomicMinF(b + 3, w); }
}

__global__ void edge_expsum(const int* __restrict__ ei, const float* __restrict__ wm,
                            const float* __restrict__ wstats,
                            const float* __restrict__ gp, const float* __restrict__ gn,
                            float* __restrict__ esum, int C) {
  long long tid = blockIdx.x * (long long)blockDim.x + threadIdx.x;
  if (tid >= (long long)N_EDGES * C) return;
  int e = (int)(tid / C);
  int c = (int)(tid % C);
  int s = ei[e];
  float w = wm[e];
  float gpc = gp[c], gnc = gn[c];
  float z = (w >= 0.f) ? w * gpc : w * gnc;
  float zm = seg_zmax(wstats + 4ll * s, gpc, gnc);
  atomicAdd(esum + (long long)s * C + c, __expf(z - zm));
}

__global__ void edge_message(const int* __restrict__ ei, const float* __restrict__ wm,
                             const float* __restrict__ wstats,
                             const float* __restrict__ gp, const float* __restrict__ gn,
                             const float* __restrict__ esum, const float* __restrict__ h,
                             float* __restrict__ agg, int C) {
  long long tid = blockIdx.x * (long long)blockDim.x + threadIdx.x;
  if (tid >= (long long)N_EDGES * C) return;
  int e = (int)(tid / C);
  int c = (int)(tid % C);
  int s = ei[e];
  int d = ei[N_EDGES + e];       // dst = edge_index[1]
  float w = wm[e];
  float gpc = gp[c], gnc = gn[c];
  float z = (w >= 0.f) ? w * gpc : w * gnc;
  float zm = seg_zmax(wstats + 4ll * s, gpc, gnc);
  float alpha = __expf(z - zm) / (esum[(long long)s * C + c] + 1e-16f);
  atomicAdd(agg + (long long)d * C + c, alpha * h[(long long)s * C + c]);
}

__global__ void elu_inplace(float* __restrict__ p, long long n) {
  long long i = blockIdx.x * (long long)blockDim.x + threadIdx.x;
  if (i < n) {
    float v = p[i];
    p[i] = v > 0.f ? v : expm1f(v);
  }
}

__global__ void log_softmax16(float* __restrict__ out) {
  int r = blockIdx.x * blockDim.x + threadIdx.x;
  if (r >= N_NODES) return;
  float* row = out + (long long)r * N_CLS;
  float m = row[0];
  #pragma unroll
  for (int i = 1; i < N_CLS; ++i) m = fmaxf(m, row[i]);
  float s = 0.f;
  #pragma unroll
  for (int i = 0; i < N_CLS; ++i) s += __expf(row[i] - m);
  float lse = m + __logf(s);
  #pragma unroll
  for (int i = 0; i < N_CLS; ++i) row[i] -= lse;
}

// ---------- orchestration ----------
extern "C" void kernel_launch(void* const* d_in, const int* in_sizes, int n_in,
                              void* d_out, int out_size, void* d_ws, size_t ws_size,
                              hipStream_t stream) {
  const float* x      = (const float*)d_in[0];
  const int*   ei     = (const int*)  d_in[1];
  const float* wm     = (const float*)d_in[2];
  const float* lin1_w = (const float*)d_in[3];
  const float* lin1_b = (const float*)d_in[4];
  const float* m1w0   = (const float*)d_in[5];
  const float* m1w1   = (const float*)d_in[6];
  // mlp1_b1 (d_in[7]) cancels in segment softmax
  const float* lin2_w = (const float*)d_in[8];
  const float* lin2_b = (const float*)d_in[9];
  const float* m2w0   = (const float*)d_in[10];
  const float* m2w1   = (const float*)d_in[11];
  // mlp2_b1 (d_in[12]) cancels as well
  float* out = (float*)d_out;

  float* ws   = (float*)d_ws;
  float* h1   = ws;                                 // N*64 (layer1 proj; reused as h2 N*16)
  float* esum = ws + (size_t)N_NODES * 64;          // N*64 (reused N*16 in layer2)
  float* agg  = esum + (size_t)N_NODES * 64;        // N*64 (layer1 aggregate -> x2)
  float* wst  = agg + (size_t)N_NODES * 64;         // N*4  (w extremes per src node)
  float* gp1  = wst + (size_t)N_NODES * 4;          // 64
  float* gn1  = gp1 + HID;                          // 64
  float* gp2  = gn1 + HID;                          // 16
  float* gn2  = gp2 + N_CLS;                        // 16

  // edge-MLP collapse vectors
  gvec_kernel<<<1, HID, 0, stream>>>(m1w0, m1w1, gp1, gn1, HID, HID);
  gvec_kernel<<<1, 32, 0, stream>>>(m2w0, m2w1, gp2, gn2, N_CLS, HID);

  // per-source w extremes (shared by both layers; depends only on w_mul/src)
  init_wstats<<<(N_NODES + 255) / 256, 256, 0, stream>>>(wst, N_NODES);
  edge_minmax<<<(N_EDGES + 255) / 256, 256, 0, stream>>>(ei, wm, wst);

  // ---- layer 1 (C = 64) ----
  {
    int waves = (N_NODES / 16) * (HID / 16);        // 25000 tiles
    gemm_wmma_f32<<<(waves + 7) / 8, 256, 0, stream>>>(x, lin1_w, lin1_b, h1,
                                                       N_NODES, F_IN, HID);
    long long n64 = (long long)N_NODES * HID;
    fill_f32<<<(unsigned)((n64 + 255) / 256), 256, 0, stream>>>(esum, 0.f, n64);
    long long ec = (long long)N_EDGES * HID;
    unsigned eb = (unsigned)((ec + 255) / 256);
    edge_expsum<<<eb, 256, 0, stream>>>(ei, wm, wst, gp1, gn1, esum, HID);
    fill_f32<<<(unsigned)((n64 + 255) / 256), 256, 0, stream>>>(agg, 0.f, n64);
    edge_message<<<eb, 256, 0, stream>>>(ei, wm, wst, gp1, gn1, esum, h1, agg, HID);
    elu_inplace<<<(unsigned)((n64 + 255) / 256), 256, 0, stream>>>(agg, n64);
  }

  // ---- layer 2 (C = 16) ----
  {
    int waves = (N_NODES / 16) * (N_CLS / 16);      // 6250 tiles
    gemm_wmma_f32<<<(waves + 7) / 8, 256, 0, stream>>>(agg, lin2_w, lin2_b, h1,
                                                       N_NODES, HID, N_CLS);
    long long n16 = (long long)N_NODES * N_CLS;
    fill_f32<<<(unsigned)((n16 + 255) / 256), 256, 0, stream>>>(esum, 0.f, n16);
    long long ec = (long long)N_EDGES * N_CLS;
    unsigned eb = (unsigned)((ec + 255) / 256);
    edge_expsum<<<eb, 256, 0, stream>>>(ei, wm, wst, gp2, gn2, esum, N_CLS);
    fill_f32<<<(unsigned)((n16 + 255) / 256), 256, 0, stream>>>(out, 0.f, n16);
    edge_message<<<eb, 256, 0, stream>>>(ei, wm, wst, gp2, gn2, esum, h1, out, N_CLS);
    log_softmax16<<<(N_NODES + 255) / 256, 256, 0, stream>>>(out);
  }
}